// RTDETRPostProcessorWithNMS_26259430048138
// MI455X (gfx1250) — compile-verified
//
#include <hip/hip_runtime.h>
#include <hip/hip_bf16.h>
#include <math.h>

namespace {
constexpr int   kN      = 1000;   // boxes per image
constexpr int   kC      = 80;     // classes
constexpr int   kPad    = 1024;   // padded (pow2) for bitonic sort
constexpr int   kTopQ   = 300;
constexpr float kNmsT   = 0.5f;
constexpr float kLogitT = -1.0986122886681098f; // logit(0.25): sigmoid(x)>0.25 <=> x>this
}

// ---------------------------------------------------------------------------
// Kernel A: per-box max+argmax over 80 logits. Wave-per-box, 8 waves/block.
// Lanes read 32 consecutive floats (perfect 128B coalescing); wave32 shfl_xor
// argmax reduction. Spreads the 41MB logits read across the whole chip.
// ---------------------------------------------------------------------------
__global__ __launch_bounds__(256)
void rtdetr_argmax(const float* __restrict__ logits,
                   float* __restrict__ wscore,
                   int*   __restrict__ wlabel,
                   int totalBoxes)
{
  const int lane = threadIdx.x & 31;
  const int wave = threadIdx.x >> 5;
  const int box  = blockIdx.x * 8 + wave;
  if (box >= totalBoxes) return;

  const float* lp = logits + (size_t)box * kC;
  float v0 = lp[lane];
  float v1 = lp[lane + 32];
  float v2 = (lane < kC - 64) ? lp[lane + 64] : -INFINITY;

  float best = v0; int bl = lane;
  if (v1 > best) { best = v1; bl = lane + 32; }
  if (v2 > best) { best = v2; bl = lane + 64; }

  #pragma unroll
  for (int d = 16; d >= 1; d >>= 1) {
    float ov = __shfl_xor(best, d);
    int   ol = __shfl_xor(bl,   d);
    if (ov > best || (ov == best && ol < bl)) { best = ov; bl = ol; }
  }
  if (lane == 0) { wscore[box] = best; wlabel[box] = bl; }
}

// ---------------------------------------------------------------------------
// Kernel B: per-image sort + NMS + compaction. One 1024-thread block per image.
// Everything LDS-resident (CDNA5: 320KB/WGP). Async-to-LDS indexed box gather.
// ---------------------------------------------------------------------------
__global__ __launch_bounds__(1024)
void rtdetr_post_fused(const float* __restrict__ logits,
                       const float* __restrict__ boxes,
                       const float* __restrict__ sizes,
                       const float* __restrict__ wscore,   // may be null
                       const int*   __restrict__ wlabel,   // may be null
                       float* __restrict__ out,
                       int nb, int fromWs)
{
  const int b    = blockIdx.x;
  const int tid  = threadIdx.x;
  const int lane = tid & 31;
  const int wave = tid >> 5;

  __shared__ float    s_key[kPad];   // sorted key: max logit (desc)
  __shared__ int      s_idx[kPad];   // payload: original box index
  __shared__ int      s_lab[kPad];   // argmax label, indexed by ORIGINAL box id
  __shared__ float4   s_box4[kPad];  // async-gather staging (cx,cy,w,h), sorted order
  __shared__ float    s_x1[kPad], s_y1[kPad], s_x2[kPad], s_y2[kPad], s_area[kPad];
  __shared__ int      s_keep[kPad];
  __shared__ unsigned s_sup[32];     // per-tile suppression bitmask rows
  __shared__ unsigned s_alivemask;   // per-tile kept-boxes bitmask
  __shared__ int      s_cnt;

  // ------------- Stage 1: sort keys (from ws, or compute in-block) ----------
  float mk; int mi;
  if (fromWs) {
    if (tid < kN) {
      mk = wscore[(size_t)b * kN + tid];
      s_lab[tid] = wlabel[(size_t)b * kN + tid];
    } else {
      mk = -INFINITY;
      s_lab[tid] = 0;
    }
  } else {
    if (tid < kN) {
      const float4* lp = (const float4*)(logits + ((size_t)b * kN + tid) * kC);
      float best = -INFINITY; int blab = 0;
      #pragma unroll
      for (int i = 0; i < kC / 4; ++i) {
        float4 v = lp[i];
        if (v.x > best) { best = v.x; blab = 4 * i + 0; }
        if (v.y > best) { best = v.y; blab = 4 * i + 1; }
        if (v.z > best) { best = v.z; blab = 4 * i + 2; }
        if (v.w > best) { best = v.w; blab = 4 * i + 3; }
      }
      mk = best; s_lab[tid] = blab;
    } else {
      mk = -INFINITY; s_lab[tid] = 0;
    }
  }
  mi = tid;

  // ------------- Stage 2: bitonic sort, descending (key, idx-tiebreak) ------
  // k = 2..32: all sub-stages are intra-wave -> pure register shfl_xor, no LDS.
  #pragma unroll
  for (int k = 2; k <= 32; k <<= 1) {
    #pragma unroll
    for (int j = k >> 1; j > 0; j >>= 1) {
      float ok = __shfl_xor(mk, j);
      int   oi = __shfl_xor(mi, j);
      bool upper     = (tid & j) != 0;
      bool descSeg   = ((tid & k) == 0);
      bool mineFirst = (mk > ok) || (mk == ok && mi < oi);
      if (mineFirst != (descSeg != upper)) { mk = ok; mi = oi; }
    }
  }
  s_key[tid] = mk; s_idx[tid] = mi;
  __syncthreads();

  // k = 64..1024: cross-wave sub-stages (j>=32) through LDS, then the 5
  // intra-wave sub-stages (j<=16) back in registers.
  for (int k = 64; k <= kPad; k <<= 1) {
    for (int j = k >> 1; j >= 32; j >>= 1) {
      int ixj = tid ^ j;
      if (ixj > tid) {
        float a = s_key[tid], c = s_key[ixj];
        int  ai = s_idx[tid], ci = s_idx[ixj];
        bool descSeg = ((tid & k) == 0);
        bool aFirst  = (a > c) || (a == c && ai < ci);
        if (aFirst != descSeg) {
          s_key[tid] = c; s_key[ixj] = a;
          s_idx[tid] = ci; s_idx[ixj] = ai;
        }
      }
      __syncthreads();
    }
    mk = s_key[tid]; mi = s_idx[tid];
    #pragma unroll
    for (int j = 16; j > 0; j >>= 1) {
      float ok = __shfl_xor(mk, j);
      int   oi = __shfl_xor(mi, j);
      bool upper     = (tid & j) != 0;
      bool descSeg   = ((tid & k) == 0);
      bool mineFirst = (mk > ok) || (mk == ok && mi < oi);
      if (mineFirst != (descSeg != upper)) { mk = ok; mi = oi; }
    }
    s_key[tid] = mk; s_idx[tid] = mi;
    __syncthreads();
  }

  // ------------- Stage 3: async-to-LDS gather of boxes in sorted order ------
  // CDNA5 async path: GLOBAL_LOAD_ASYNC_TO_LDS_B128 (GVS mode), memory -> LDS
  // directly (no VGPR round trip), tracked with ASYNCcnt.
  if (tid < kN) {
    int src = s_idx[tid];
    unsigned     vOff   = (unsigned)src * 16u;               // per-lane byte offset
    const float* sbase  = boxes + (size_t)b * kN * 4;        // uniform base (SGPR pair)
    unsigned     ldsAdr = (unsigned)(size_t)(&s_box4[tid]);  // low 32 bits = LDS addr
    asm volatile("global_load_async_to_lds_b128 %0, %1, %2"
                 :: "v"(ldsAdr), "v"(vOff), "s"(sbase)
                 : "memory");
  }
#if __has_builtin(__builtin_amdgcn_s_wait_asynccnt)
  __builtin_amdgcn_s_wait_asynccnt(0);
#else
  asm volatile("s_wait_asynccnt 0" ::: "memory");
#endif
  __syncthreads();

  const float sw = sizes[b * 2 + 0];
  const float sh = sizes[b * 2 + 1];

  if (tid < kN) {
    float4 bx = s_box4[tid];
    float x1 = (bx.x - 0.5f * bx.z) * sw;
    float y1 = (bx.y - 0.5f * bx.w) * sh;
    float x2 = (bx.x + 0.5f * bx.z) * sw;
    float y2 = (bx.y + 0.5f * bx.w) * sh;
    s_x1[tid] = x1; s_y1[tid] = y1; s_x2[tid] = x2; s_y2[tid] = y2;
    s_area[tid] = (x2 - x1) * (y2 - y1);
    s_keep[tid] = (s_key[tid] > kLogitT) ? 1 : 0;   // valid = score > 0.25
  } else {
    s_x1[tid] = 0.f; s_y1[tid] = 0.f; s_x2[tid] = 0.f; s_y2[tid] = 0.f;
    s_area[tid] = 0.f;
    s_keep[tid] = 0;
  }
  __syncthreads();

  // ------------- Stage 4: tiled greedy NMS (bitmask formulation) ------------
  // Per 32-box tile: (a) all 1024 threads compute the tile's 32x32 IoU matrix
  // (wave = row, lane = col), ballot -> one suppression word per row;
  // (b) wave 0 resolves the greedy recurrence with pure bit ops (identical
  // semantics to the reference fori_loop); (c) later boxes test only the
  // tile's surviving boxes via ffs bit iteration.
  for (int tile = 0; tile < kPad / 32; ++tile) {
    const int base = tile * 32;

    {   // (a) 32x32 suppression matrix
      int i = base + wave;
      int j = base + lane;
      float iw = fminf(s_x2[i], s_x2[j]) - fmaxf(s_x1[i], s_x1[j]);
      float ih = fminf(s_y2[i], s_y2[j]) - fmaxf(s_y1[i], s_y1[j]);
      iw = fmaxf(iw, 0.f); ih = fmaxf(ih, 0.f);
      float inter = iw * ih;
      float iou = inter / (s_area[i] + s_area[j] - inter + 1e-9f);
      unsigned m = (unsigned)__ballot((lane > wave) && (iou > kNmsT));
      if (lane == 0) s_sup[wave] = m;
    }
    __syncthreads();

    if (wave == 0) {   // (b) greedy resolve over bitmasks
      unsigned km = (unsigned)__ballot(s_keep[base + lane] != 0);
      unsigned alive = km, donem = 0;
      while (true) {
        unsigned cand = alive & ~donem;
        if (cand == 0) break;
        int p = __ffs((int)cand) - 1;   // lowest-index alive, unprocessed => kept
        donem |= (1u << p);
        alive &= ~s_sup[p];             // s_sup[p] holds only bits > p
      }
      s_keep[base + lane] = (alive >> lane) & 1;
      if (lane == 0) s_alivemask = alive;
    }
    __syncthreads();

    {   // (c) suppress all later boxes against the tile's kept boxes
      int j = base + 32 + tid;
      if (j < kPad && s_keep[j]) {
        unsigned m = s_alivemask;
        float jx1 = s_x1[j], jy1 = s_y1[j], jx2 = s_x2[j], jy2 = s_y2[j], ja = s_area[j];
        int alive = 1;
        while (m) {
          int p = __ffs((int)m) - 1; m &= (m - 1);
          int i = base + p;
          float iw = fminf(jx2, s_x2[i]) - fmaxf(jx1, s_x1[i]);
          float ih = fminf(jy2, s_y2[i]) - fmaxf(jy1, s_y1[i]);
          iw = fmaxf(iw, 0.f); ih = fmaxf(ih, 0.f);
          float inter = iw * ih;
          float iou = inter / (ja + s_area[i] - inter + 1e-9f);
          if (iou > kNmsT) { alive = 0; break; }
        }
        if (!alive) s_keep[j] = 0;
      }
    }
    __syncthreads();
  }

  // ------------- Stage 5: ballot-based compaction + output ------------------
  // Output layout (flat concat, return order):
  //   labels[nb*300] | boxes[nb*300*4] | scores[nb*300]
  float* out_l = out;
  float* out_b = out + (size_t)nb * kTopQ;
  float* out_s = out + (size_t)nb * kTopQ * 5;

  if (wave == 0) {
    int running = 0;
    for (int c = 0; c < kPad / 32; ++c) {
      int slot = c * 32 + lane;
      int f = s_keep[slot];
      unsigned mask  = (unsigned)__ballot(f != 0);
      int      below = __popc(mask & ((1u << lane) - 1u));
      int      pos   = running + below;
      if (f && pos < kTopQ) {
        int   orig  = s_idx[slot];
        float lg    = s_key[slot];
        float score = 1.0f / (1.0f + __expf(-lg));
        out_l[(size_t)b * kTopQ + pos] = (float)s_lab[orig];
        out_s[(size_t)b * kTopQ + pos] = score;
        float* ob = out_b + ((size_t)b * kTopQ + pos) * 4;
        ob[0] = s_x1[slot]; ob[1] = s_y1[slot]; ob[2] = s_x2[slot]; ob[3] = s_y2[slot];
      }
      running += __popc(mask);
    }
    if (lane == 0) s_cnt = (running < kTopQ) ? running : kTopQ;
  }
  __syncthreads();

  // Zero-fill unused slots (every output element written each call).
  for (int pos = s_cnt + tid; pos < kTopQ; pos += 1024) {
    out_l[(size_t)b * kTopQ + pos] = 0.f;
    out_s[(size_t)b * kTopQ + pos] = 0.f;
    float* ob = out_b + ((size_t)b * kTopQ + pos) * 4;
    ob[0] = 0.f; ob[1] = 0.f; ob[2] = 0.f; ob[3] = 0.f;
  }
}

extern "C" void kernel_launch(void* const* d_in, const int* in_sizes, int n_in,
                              void* d_out, int out_size, void* d_ws, size_t ws_size,
                              hipStream_t stream) {
  (void)n_in; (void)out_size;
  const float* logits = (const float*)d_in[0];
  const float* boxes  = (const float*)d_in[1];
  const float* sizes  = (const float*)d_in[2];
  float* out = (float*)d_out;
  const int nb = in_sizes[2] / 2;   // batch = 128

  const size_t needWs = (size_t)nb * kN * (sizeof(float) + sizeof(int));
  const bool useWs = (d_ws != nullptr) && (ws_size >= needWs);

  float* wscore = (float*)d_ws;
  int*   wlabel = (int*)((char*)d_ws + (size_t)nb * kN * sizeof(float));

  if (useWs) {
    const int total   = nb * kN;
    const int blocksA = (total + 7) / 8;   // 8 waves (boxes) per 256-thread block
    hipLaunchKernelGGL(rtdetr_argmax, dim3(blocksA), dim3(256), 0, stream,
                       logits, wscore, wlabel, total);
  }
  hipLaunchKernelGGL(rtdetr_post_fused, dim3(nb), dim3(1024), 0, stream,
                     logits, boxes, sizes,
                     useWs ? wscore : nullptr,
                     useWs ? wlabel : nullptr,
                     out, nb, useWs ? 1 : 0);
}